// CTCLayer_14353780703879
// MI455X (gfx1250) — compile-verified
//
#include <hip/hip_runtime.h>
#include <hip/hip_bf16.h>
#include <cstddef>

#define B_  256
#define T_  256
#define C_  1024
#define L_  64
#define S_  129            // 2L+1
#define SP_ 132            // padded emit stride
#define BLANK_ 1023
#define NEGV (-1e30f)
#define EPS_ (1e-7f)
#define CEPS_ (1024.0f * 1e-7f)
#define RTH_ 160           // recursion threads (5 waves, >= S_)

typedef __attribute__((ext_vector_type(2))) float v2f;
typedef __attribute__((ext_vector_type(8))) float v8f;

// ---------------------------------------------------------------------------
// Kernel 1: denom[b,t] = sum_c y_pred[b,t,c]  via V_WMMA_F32_16X16X4_F32.
// A = ones(16x4)  =>  D[m][n] = C[m][n] + sum_k B[k][n]; accumulate over C.
// B layout (32-bit 4x16): lane l (n = l&15, h = l>>4) holds K = 2h (VGPR0)
// and K = 2h+1 (VGPR1) -> a single 8-byte load per lane per WMMA, zero waste.
// One wave handles 16 (b,t) rows; 256 WMMAs sweep C=1024.
// ---------------------------------------------------------------------------
__global__ __launch_bounds__(256) void k_rowsum_wmma(const float* __restrict__ yp,
                                                     float* __restrict__ logden) {
  const int tid  = threadIdx.x;
  const int wave = tid >> 5;
  const int lane = tid & 31;
  const int n    = lane & 15;   // which row (N index)
  const int half = lane >> 4;   // K half
  const int row  = blockIdx.x * 128 + wave * 16 + n;   // row = b*T + t

  const float* p = yp + (size_t)row * C_ + (half << 1);

  v2f aOnes; aOnes[0] = 1.0f; aOnes[1] = 1.0f;
  v8f acc = {};

#pragma unroll 4
  for (int c = 0; c < C_; c += 4) {
    v2f bv = *(const v2f*)(p + c);   // K = c+2h, c+2h+1 for column n
    acc = __builtin_amdgcn_wmma_f32_16x16x4_f32(
        /*neg_a=*/false, aOnes, /*neg_b=*/false, bv,
        /*c_mod=*/(short)0, acc, /*reuse_a=*/false, /*reuse_b=*/false);
  }
  // Every D row equals the column sum; lane n (lanes 0..15), VGPR0 = D[0][n].
  if (half == 0) logden[row] = __logf(acc[0] + CEPS_);
}

// ---------------------------------------------------------------------------
// Kernel 2a: staged emit gather.  emit[b,t,j] = log(y[b,t,ext_j]+EPS) - logden
// One 160-thread block per (t, b); fully parallel, hides gather latency.
// ---------------------------------------------------------------------------
__global__ __launch_bounds__(RTH_) void k_gather(const long long* __restrict__ yt,
                                                 const float* __restrict__ yp,
                                                 const float* __restrict__ logden,
                                                 float* __restrict__ emit) {
  const int t = blockIdx.x;
  const int b = blockIdx.y;
  const int j = threadIdx.x;
  if (j >= S_) return;
  int extj = BLANK_;
  if (j & 1) extj = (int)yt[(size_t)b * L_ + (j >> 1)];
  const float ld = logden[(size_t)b * T_ + t];
  const float v  = yp[((size_t)b * T_ + t) * C_ + extj];
  emit[((size_t)b * T_ + t) * SP_ + j] = __logf(v + EPS_) - ld;
}

// shared logsumexp3 step
__device__ __forceinline__ float lse3(float a1, float a2, float a3) {
  float m = fmaxf(a1, fmaxf(a2, a3));
  return m + __logf(__expf(a1 - m) + __expf(a2 - m) + __expf(a3 - m));
}

// ---------------------------------------------------------------------------
// Kernel 3 (path A): alpha recursion over staged emit. One block per batch,
// LDS ping-pong alpha, one barrier per timestep, prefetch of emit t+2.
// ---------------------------------------------------------------------------
__global__ __launch_bounds__(RTH_) void k_ctc_staged(const long long* __restrict__ yt,
                                                     const float* __restrict__ emit,
                                                     float* __restrict__ out) {
  const int b = blockIdx.x;
  const int j = threadIdx.x;
  __shared__ float A[2][RTH_];

  const bool active = (j < S_);
  bool skip = false;
  if (active && (j & 1) && j >= 3) {
    const long long* lb = yt + (size_t)b * L_;
    skip = (lb[j >> 1] != lb[(j >> 1) - 1]);
  }
  const float* E = emit + (size_t)b * T_ * SP_;

  float a0 = (active && j < 2) ? E[j] : NEGV;
  A[0][j] = a0;
  __syncthreads();

  int cur = 0;
  for (int t = 1; t < T_; ++t) {
    float e = active ? E[(size_t)t * SP_ + j] : NEGV;
    int tp = (t + 2 < T_) ? (t + 2) : t;
    __builtin_prefetch(E + (size_t)tp * SP_ + j, 0, 1);   // global_prefetch_b8
    float a1 = A[cur][j];
    float a2 = (j >= 1) ? A[cur][j - 1] : NEGV;
    float a3 = skip ? A[cur][j - 2] : NEGV;
    float an = active ? (lse3(a1, a2, a3) + e) : NEGV;
    A[cur ^ 1][j] = an;
    __syncthreads();
    cur ^= 1;
  }
  if (j == 0) {
    float x = A[cur][S_ - 2], y = A[cur][S_ - 1];
    float m = fmaxf(x, y);
    out[b] = -(m + __logf(__expf(x - m) + __expf(y - m)));
  }
}

// block-wide log(sum_c row[c] + C*EPS) for the no-workspace fallback
__device__ __forceinline__ float block_logden(const float* __restrict__ row,
                                              int j, float* sden) {
  if (j == 0) *sden = 0.0f;
  __syncthreads();
  float p = 0.0f;
  for (int c = j; c < C_; c += RTH_) p += row[c];
  atomicAdd(sden, p);                 // ds_add_f32
  __syncthreads();
  float r = __logf(*sden + CEPS_);
  __syncthreads();
  return r;
}

// ---------------------------------------------------------------------------
// Kernel 3' (paths B/C): recursion with inline gather (prefetched one step
// ahead in registers). logden may be null -> compute denom per step in-block.
// ---------------------------------------------------------------------------
__global__ __launch_bounds__(RTH_) void k_ctc_inline(const long long* __restrict__ yt,
                                                     const float* __restrict__ yp,
                                                     const float* __restrict__ logden,
                                                     float* __restrict__ out) {
  const int b = blockIdx.x;
  const int j = threadIdx.x;
  __shared__ float A[2][RTH_];
  __shared__ float sden;

  const bool active = (j < S_);
  const bool haveLd = (logden != nullptr);
  int  extj = BLANK_;
  bool skip = false;
  if (active && (j & 1)) {
    const long long* lb = yt + (size_t)b * L_;
    long long v = lb[j >> 1];
    extj = (int)v;
    if (j >= 3) skip = (v != lb[(j >> 1) - 1]);
  }
  const float* Y = yp + (size_t)b * T_ * C_;

  float ycur = active ? Y[extj] : 1.0f;                        // t = 0 gather
  float ld   = haveLd ? logden[(size_t)b * T_] : block_logden(Y, j, &sden);
  float a0   = (active && j < 2) ? (__logf(ycur + EPS_) - ld) : NEGV;
  A[0][j] = a0;
  __syncthreads();

  float ynext = active ? Y[(size_t)C_ + extj] : 1.0f;          // prefetch t=1
  int cur = 0;
  for (int t = 1; t < T_; ++t) {
    float y = ynext;
    if (t + 1 < T_) ynext = active ? Y[(size_t)(t + 1) * C_ + extj] : 1.0f;
    float ldt = haveLd ? logden[(size_t)b * T_ + t]
                       : block_logden(Y + (size_t)t * C_, j, &sden);
    float a1 = A[cur][j];
    float a2 = (j >= 1) ? A[cur][j - 1] : NEGV;
    float a3 = skip ? A[cur][j - 2] : NEGV;
    float e  = __logf(y + EPS_) - ldt;
    float an = active ? (lse3(a1, a2, a3) + e) : NEGV;
    A[cur ^ 1][j] = an;
    __syncthreads();
    cur ^= 1;
  }
  if (j == 0) {
    float x = A[cur][S_ - 2], y2 = A[cur][S_ - 1];
    float m = fmaxf(x, y2);
    out[b] = -(m + __logf(__expf(x - m) + __expf(y2 - m)));
  }
}

// ---------------------------------------------------------------------------
extern "C" void kernel_launch(void* const* d_in, const int* in_sizes, int n_in,
                              void* d_out, int out_size, void* d_ws, size_t ws_size,
                              hipStream_t stream) {
  (void)in_sizes; (void)n_in; (void)out_size;
  const long long* y_true = (const long long*)d_in[0];   // (B, L) int64
  const float*     y_pred = (const float*)d_in[1];       // (B, T, C) f32
  float*           out    = (float*)d_out;               // (B, 1) f32

  const size_t ldBytes   = (size_t)B_ * T_ * sizeof(float);                 // 256 KB
  const size_t emitBytes = ((size_t)B_ * T_ * SP_ + 64) * sizeof(float);    // ~34.6 MB

  float* logden = (float*)d_ws;
  float* emit   = (float*)((char*)d_ws + ldBytes);

  const bool haveLd   = ws_size >= ldBytes;
  const bool haveEmit = ws_size >= ldBytes + emitBytes;

  if (haveLd) {
    // 65536 rows, 16 rows/wave, 8 waves/block -> 512 blocks
    k_rowsum_wmma<<<512, 256, 0, stream>>>(y_pred, logden);
  }
  if (haveEmit) {
    k_gather<<<dim3(T_, B_), RTH_, 0, stream>>>(y_true, y_pred, logden, emit);
    k_ctc_staged<<<B_, RTH_, 0, stream>>>(y_true, emit, out);
  } else {
    k_ctc_inline<<<B_, RTH_, 0, stream>>>(y_true, y_pred,
                                          haveLd ? logden : nullptr, out);
  }
}